// GELU65_17566416240671
// MI455X (gfx1250) — compile-verified
//
#include <hip/hip_runtime.h>
#include <math.h>

// Problem constants (from reference): x [B, T, D] float32
#define BB 8
#define TT 4096
#define DD 1024
#define LL 16                 // chunk length for the parallel scan
#define NCHUNK (TT / LL)      // 256
#define EPSV 1e-5f

typedef int v4i_t __attribute__((ext_vector_type(4)));
typedef __attribute__((address_space(1))) v4i_t* as1_v4i;
typedef __attribute__((address_space(3))) v4i_t* as3_v4i;

__device__ __forceinline__ float tanh_fast(float z) {
#if defined(__gfx1250__) && __has_builtin(__builtin_amdgcn_tanhf)
    return __builtin_amdgcn_tanhf(z);   // V_TANH_F32 (CDNA5 TRANS op)
#else
    return tanhf(z);
#endif
}

__device__ __forceinline__ float gelu_f(float x) {
    const float c = 0.79788456080286535588f; // sqrt(2/pi)
    float x3 = x * x * x;
    float t = tanh_fast(c * (x + 0.044715f * x3));
    return 0.5f * x * (1.0f + t);
}

__device__ __forceinline__ float softplus_f(float x) {
    return (x > 20.0f) ? x : log1pf(expf(x));
}

__device__ __forceinline__ float sigmoid_f(float x) {
    return 1.0f / (1.0f + expf(-x));
}

// ---------------------------------------------------------------------------
// K1: per-(b, chunk, d) affine chunk contribution of the linear Ca recurrence
//     S = sum_{i=0..L-1} d^{L-1-i} * (1-d) * |gelu(x_{t0+i})|
// ---------------------------------------------------------------------------
__global__ __launch_bounds__(256)
void k_chunk_sums(const float* __restrict__ x,
                  const float* __restrict__ logit_d_ca,
                  float* __restrict__ S) {
    int gid   = blockIdx.x * 256 + threadIdx.x;      // 0 .. B*NCHUNK*D-1
    int d     = gid & (DD - 1);
    int chunk = (gid >> 10) & (NCHUNK - 1);
    int b     = gid >> 18;                           // / (D*NCHUNK)

    float dca = sigmoid_f(logit_d_ca[0]);
    float w   = 1.0f - dca;

    const float* xp = x + ((size_t)(b * TT + chunk * LL)) * DD + d;
    float s = 0.0f;
#pragma unroll
    for (int i = 0; i < LL; ++i) {
        float g = gelu_f(xp[(size_t)i * DD]);
        s = dca * s + w * fabsf(g);
    }
    S[gid] = s;
}

// ---------------------------------------------------------------------------
// K2: sequential combine across chunks (tiny): Ca at each chunk start
// ---------------------------------------------------------------------------
__global__ __launch_bounds__(256)
void k_chunk_scan(const float* __restrict__ S,
                  const float* __restrict__ ema_ca,
                  const float* __restrict__ logit_d_ca,
                  float* __restrict__ CaStart) {
    int tid = blockIdx.x * 256 + threadIdx.x;        // 0 .. B*D-1
    int d   = tid & (DD - 1);
    int b   = tid >> 10;

    float dca = sigmoid_f(logit_d_ca[0]);
    float dl  = dca * dca;   // ^2
    dl = dl * dl;            // ^4
    dl = dl * dl;            // ^8
    dl = dl * dl;            // ^16 == dca^LL

    float ca = ema_ca[d];
    for (int c = 0; c < NCHUNK; ++c) {
        size_t idx = ((size_t)(b * NCHUNK + c)) * DD + d;
        CaStart[idx] = ca;
        ca = dl * ca + S[idx];
    }
}

// ---------------------------------------------------------------------------
// K3: fused gate replay + D-mean normalize + output
//     one block = (b, chunk): 1024 threads (one per d), 16 time steps
// ---------------------------------------------------------------------------
__global__ __launch_bounds__(1024)
void k_main(const float* __restrict__ x,
            const float* __restrict__ log_beta_raw,
            const float* __restrict__ logit_d_ca,
            const float* __restrict__ CaStart,
            float* __restrict__ out) {
    __shared__ float xs[LL * DD];       // 64 KB x-tile
    __shared__ float partial[LL * 32];  // per-(t, wave) partial sums
    __shared__ float inv_mean[LL];      // per-t 1/(mean_D(gate)+eps)

    int tid   = threadIdx.x;            // == d
    int blk   = blockIdx.x;             // b*NCHUNK + chunk
    int b     = blk >> 8;               // / NCHUNK
    int chunk = blk & (NCHUNK - 1);
    int t0    = chunk * LL;

    const float* xbase = x + ((size_t)(b * TT + t0)) * DD;

    // ---- stage 16x1024 f32 tile into LDS (async-to-LDS on gfx1250) ----
#if defined(__gfx1250__) && __has_builtin(__builtin_amdgcn_global_load_async_to_lds_b128) && __has_builtin(__builtin_amdgcn_s_wait_asynccnt)
#pragma unroll
    for (int j = 0; j < 4; ++j) {
        int idx = j * 1024 + tid;       // float4 index, coalesced per wave
        __builtin_amdgcn_global_load_async_to_lds_b128(
            (as1_v4i)(xbase + (size_t)idx * 4),
            (as3_v4i)(&xs[idx * 4]),
            0, 0);
    }
    __builtin_amdgcn_s_wait_asynccnt(0);
    __syncthreads();
#else
#pragma unroll
    for (int j = 0; j < 4; ++j) {
        int idx = j * 1024 + tid;
        ((float4*)xs)[idx] = ((const float4*)xbase)[idx];
    }
    __syncthreads();
#endif

    float beta = softplus_f(log_beta_raw[0]);
    float dca  = sigmoid_f(logit_d_ca[0]);
    float w    = 1.0f - dca;

    float ca = CaStart[((size_t)blk) * DD + tid];

    float gate[LL];
#pragma unroll
    for (int i = 0; i < LL; ++i) {
        float xv = xs[i * DD + tid];
        float g  = gelu_f(xv);
        gate[i]  = 1.0f / (1.0f + beta * ca);   // gate uses pre-update Ca
        ca = dca * ca + w * fabsf(g);
    }

    // ---- reduce gate over D for each t: wave32 butterfly, then cross-wave ----
    int lane = tid & 31;
    int wave = tid >> 5;
#pragma unroll
    for (int i = 0; i < LL; ++i) {
        float r = gate[i];
#pragma unroll
        for (int m = 16; m >= 1; m >>= 1) r += __shfl_xor(r, m, 32);
        if (lane == 0) partial[i * 32 + wave] = r;
    }
    __syncthreads();
    if (wave < LL) {
        float r = partial[wave * 32 + lane];
#pragma unroll
        for (int m = 16; m >= 1; m >>= 1) r += __shfl_xor(r, m, 32);
        if (lane == 0) inv_mean[wave] = 1.0f / (r * (1.0f / (float)DD) + EPSV);
    }
    __syncthreads();

    // ---- normalize and write: out = gelu(x) * gate / (mean_D(gate) + eps) ----
    float* outp = out + ((size_t)(b * TT + t0)) * DD + tid;
#pragma unroll
    for (int i = 0; i < LL; ++i) {
        float xv = xs[i * DD + tid];
        float g  = gelu_f(xv);
        outp[(size_t)i * DD] = g * gate[i] * inv_mean[i];
    }
}

// ---------------------------------------------------------------------------
extern "C" void kernel_launch(void* const* d_in, const int* in_sizes, int n_in,
                              void* d_out, int out_size, void* d_ws, size_t ws_size,
                              hipStream_t stream) {
    // setup_inputs order: x, log_beta_raw, logit_d_ca, logit_decay (unused), ema_ca
    const float* x            = (const float*)d_in[0];
    const float* log_beta_raw = (const float*)d_in[1];
    const float* logit_d_ca   = (const float*)d_in[2];
    const float* ema_ca       = (const float*)d_in[4];
    float* out = (float*)d_out;

    // workspace: S [B*NCHUNK*D] f32 (8 MB) + CaStart [B*NCHUNK*D] f32 (8 MB)
    float* S       = (float*)d_ws;
    float* CaStart = S + (size_t)BB * NCHUNK * DD;

    const int n1 = BB * NCHUNK * DD;   // 2,097,152
    k_chunk_sums<<<n1 / 256, 256, 0, stream>>>(x, logit_d_ca, S);

    const int n2 = BB * DD;            // 8192
    k_chunk_scan<<<n2 / 256, 256, 0, stream>>>(S, ema_ca, logit_d_ca, CaStart);

    k_main<<<BB * NCHUNK, 1024, 0, stream>>>(x, log_beta_raw, logit_d_ca, CaStart, out);
}